// InterMessage_68049461838554
// MI455X (gfx1250) — compile-verified
//
#include <hip/hip_runtime.h>

typedef __attribute__((ext_vector_type(2))) float v2f;
typedef __attribute__((ext_vector_type(8))) float v8f;

#define DIM 256
#define LDS_STRIDE 260  // 256 + 4 pad: bank = (4*row + k) % 64, conflict-free

// ---------------- utility kernels ----------------

__global__ void fill_zero_kernel(float* __restrict__ p, int n) {
    int i = blockIdx.x * blockDim.x + threadIdx.x;
    if (i < n) p[i] = 0.0f;
}

// Pair-interleaved transpose:
//   Bp[(k>>1)*512 + n*2 + (k&1)] = W[n*256 + k]
// i.e. Bp row j holds float2{ B[2j][n], B[2j+1][n] } for n = 0..255,
// where B[k][n] = W[n][k]. One WMMA B fragment pair = one b64 load per lane.
__global__ void transpose_pair_kernel(const float* __restrict__ W, float* __restrict__ Bp) {
    int k = blockIdx.x;      // 0..255
    int n = threadIdx.x;     // 0..255
    Bp[(k >> 1) * (2 * DIM) + n * 2 + (k & 1)] = W[n * DIM + k];
}

// one thread per float4 chunk of an edge row: 64 chunks/edge
__global__ void scatter_add_kernel(const float4* __restrict__ from,
                                   const int* __restrict__ idx,
                                   float* __restrict__ sums, int nItems) {
    int i = blockIdx.x * blockDim.x + threadIdx.x;
    if (i >= nItems) return;
    int e = i >> 6;
    int c = i & 63;
    int node = idx[e];
    float4 v = from[(size_t)e * 64 + c];
    float* dst = sums + (size_t)node * DIM + c * 4;
    atomicAdd(dst + 0, v.x);
    atomicAdd(dst + 1, v.y);
    atomicAdd(dst + 2, v.z);
    atomicAdd(dst + 3, v.w);
}

__global__ void count_kernel(const int* __restrict__ idx, float* __restrict__ cnt, int nEdges) {
    int e = blockIdx.x * blockDim.x + threadIdx.x;
    if (e < nEdges) atomicAdd(&cnt[idx[e]], 1.0f);
}

// cnt[i] <- 1 / max(cnt[i], 1)
__global__ void invert_count_kernel(float* __restrict__ cnt, int n) {
    int i = blockIdx.x * blockDim.x + threadIdx.x;
    if (i < n) cnt[i] = 1.0f / fmaxf(cnt[i], 1.0f);
}

// ---------------- fp32 WMMA GEMM: Out = relu((A*rowScale) @ B + bias) ----------------
// A:  [M x 256] row-major, rowScale: [M] or nullptr.
// Bp: pair-interleaved B (see transpose_pair_kernel), bias: [256], Out: [M x 256].
// Block = 256 threads (8 waves). Block owns 16 rows x 256 cols; wave w owns cols [32w, 32w+32).
__global__ void __launch_bounds__(256)
gemm_relu_wmma_kernel(const float* __restrict__ A,
                      const float* __restrict__ rowScale,
                      const float* __restrict__ Bp,
                      const float* __restrict__ bias,
                      float* __restrict__ Out, int M) {
    __shared__ float ldsA[16 * LDS_STRIDE];

    const int m0 = blockIdx.x * 16;
    const int t  = threadIdx.x;

    // Cooperative, coalesced load of the 16x256 A strip, fused with row scaling.
    for (int i = 0; i < 16; ++i) {
        int row = m0 + i;
        float v = 0.0f;
        if (row < M) {
            float s = rowScale ? rowScale[row] : 1.0f;
            v = A[(size_t)row * DIM + t] * s;
        }
        ldsA[i * LDS_STRIDE + t] = v;
    }
    __syncthreads();

    const int lane = t & 31;
    const int wave = t >> 5;
    const int n0   = wave * 32;
    const int mrow = lane & 15;          // A fragment row
    const int koff = (lane >> 4) * 2;    // A/B fragment K sub-offset (0 or 2)
    const int ncol = lane & 15;          // B/C fragment column

    // Per-lane base pointers into the pair-interleaved B:
    //   fragment for k needs pair row j = (k + koff) / 2, column n.
    const float* bbase0 = Bp + (size_t)(koff >> 1) * (2 * DIM) + (n0 + ncol) * 2;
    const float* bbase1 = bbase0 + 32;   // columns n0+16..n0+31

    v8f acc0 = {0.f, 0.f, 0.f, 0.f, 0.f, 0.f, 0.f, 0.f};
    v8f acc1 = {0.f, 0.f, 0.f, 0.f, 0.f, 0.f, 0.f, 0.f};

    for (int k = 0; k < DIM; k += 4) {
        // A 16x4 fp32 fragment: lane<16 -> K=k..k+1, lane>=16 -> K=k+2..k+3
        v2f a;
        a.x = ldsA[mrow * LDS_STRIDE + k + koff];
        a.y = ldsA[mrow * LDS_STRIDE + k + koff + 1];

        // B 4x16 fp32 fragments: one b64 per lane covers both K-rows of the pair.
        v2f b0 = *(const v2f*)(bbase0 + (size_t)(k >> 1) * (2 * DIM));
        v2f b1 = *(const v2f*)(bbase1 + (size_t)(k >> 1) * (2 * DIM));

        acc0 = __builtin_amdgcn_wmma_f32_16x16x4_f32(false, a, false, b0,
                                                     (short)0, acc0, false, false);
        acc1 = __builtin_amdgcn_wmma_f32_16x16x4_f32(false, a, false, b1,
                                                     (short)0, acc1, false, false);
    }

    // Epilogue: C/D layout — VGPR r: lanes 0-15 -> M=r, lanes 16-31 -> M=r+8.
    const int mbase = m0 + (lane >> 4) * 8;
    const float bia0 = bias[n0 + ncol];
    const float bia1 = bias[n0 + 16 + ncol];
    for (int r = 0; r < 8; ++r) {
        int row = mbase + r;
        if (row < M) {
            float x0 = acc0[r] + bia0;
            float x1 = acc1[r] + bia1;
            Out[(size_t)row * DIM + n0 + ncol]      = x0 > 0.0f ? x0 : 0.0f;
            Out[(size_t)row * DIM + n0 + 16 + ncol] = x1 > 0.0f ? x1 : 0.0f;
        }
    }
}

// ---------------- launcher ----------------

extern "C" void kernel_launch(void* const* d_in, const int* in_sizes, int n_in,
                              void* d_out, int out_size, void* d_ws, size_t ws_size,
                              hipStream_t stream) {
    const float* from = (const float*)d_in[0];   // [nEdges, 256]
    const int*   idx  = (const int*)d_in[1];     // [nEdges] (int32)
    const float* W1   = (const float*)d_in[3];   // [256, 256]
    const float* b1   = (const float*)d_in[4];   // [256]
    const float* W2   = (const float*)d_in[5];   // [256, 256]
    const float* b2   = (const float*)d_in[6];   // [256]
    float* out = (float*)d_out;                  // [M, 256]

    const int M      = out_size / DIM;           // n_nodes (50000)
    const int nEdges = in_sizes[0] / DIM;        // 800000

    // workspace layout: h [M*256] | cnt [M] | W1p [256*256] | W2p [256*256]
    char*  ws  = (char*)d_ws;
    float* h   = (float*)ws;
    float* cnt = (float*)(ws + (size_t)M * DIM * sizeof(float));
    float* W1p = cnt + M;
    float* W2p = W1p + DIM * DIM;

    const int B = 256;

    // 1) zero sums (accumulated in d_out) and counts
    fill_zero_kernel<<<(M * DIM + B - 1) / B, B, 0, stream>>>(out, M * DIM);
    fill_zero_kernel<<<(M + B - 1) / B, B, 0, stream>>>(cnt, M);

    // 2) pair-interleaved transposed weights (one b64/lane B-fragment loads in GEMM)
    transpose_pair_kernel<<<DIM, DIM, 0, stream>>>(W1, W1p);
    transpose_pair_kernel<<<DIM, DIM, 0, stream>>>(W2, W2p);

    // 3) scatter-add features + edge counts
    int nItems = nEdges * (DIM / 4);
    scatter_add_kernel<<<(nItems + B - 1) / B, B, 0, stream>>>((const float4*)from, idx, out, nItems);
    count_kernel<<<(nEdges + B - 1) / B, B, 0, stream>>>(idx, cnt, nEdges);

    // 4) counts -> inverse (mean divide fused into GEMM1's A load)
    invert_count_kernel<<<(M + B - 1) / B, B, 0, stream>>>(cnt, M);

    // 5) h = relu(mean @ W1^T + b1)     (A = sums in d_out, scaled by cnt)
    int mtiles = (M + 15) / 16;
    gemm_relu_wmma_kernel<<<mtiles, 256, 0, stream>>>(out, cnt, W1p, b1, h, M);

    // 6) out = relu(h @ W2^T + b2)
    gemm_relu_wmma_kernel<<<mtiles, 256, 0, stream>>>(h, nullptr, W2p, b2, out, M);
}